// LightGCN_5884105195911
// MI455X (gfx1250) — compile-verified
//
#include <hip/hip_runtime.h>
#include <hip/hip_bf16.h>
#include <cstdint>

// LightGCN propagation on MI455X (gfx1250).
//
// Roofline: 1.84 GFLOP vs ~7.4 GB of L2-resident gather/scatter traffic
// (AI ~ 0.25 FLOP/B) -> memory/atomic bound, WMMA not applicable.
// Strategy: build CSR once (kills 922M f32 atomic-RMWs), then 3 gather-based
// SpMM passes. Node features (2 x 38.4 MB ping-pong + 38.4 MB acc) stay in
// the 192 MB L2. Edge (col,val) streams are staged into per-wave LDS buffers
// with CDNA5 async global->LDS DMA, double-buffered via s_wait_asynccnt.

#define N_USERS   100000
#define N_ITEMS   50000
#define N_NODES   150000
#define EMB       64
#define N_LAYERS  3
#define TOTAL     (N_NODES * EMB)        // 9,600,000 floats
#define BLOCK     256                    // 8 wave32 per block
#define WPB       (BLOCK / 32)           // waves per block

// -------- CDNA5 async global->LDS staging (probe via __has_builtin) --------
#if __has_builtin(__builtin_amdgcn_global_load_async_to_lds_b32)
#define HAVE_ASYNC_LDS 1
typedef __attribute__((address_space(1))) int gbl_int;   // global (AS1)
typedef __attribute__((address_space(3))) int lds_int;   // LDS    (AS3)

__device__ __forceinline__ gbl_int* as_gbl(const void* p) {
  return (gbl_int*)(uintptr_t)p;
}
__device__ __forceinline__ lds_int* as_lds(void* p) {
  // generic shared pointer: low 32 bits are the LDS byte offset (aperture rule)
  return (lds_int*)(uint32_t)(uintptr_t)p;
}
#endif

__device__ __forceinline__ void wait_async_le2() {
#if defined(HAVE_ASYNC_LDS)
#if __has_builtin(__builtin_amdgcn_s_wait_asynccnt)
  __builtin_amdgcn_s_wait_asynccnt(2);
#else
  asm volatile("s_wait_asynccnt 0x2" ::: "memory");
#endif
#endif
}
__device__ __forceinline__ void wait_async_le0() {
#if defined(HAVE_ASYNC_LDS)
#if __has_builtin(__builtin_amdgcn_s_wait_asynccnt)
  __builtin_amdgcn_s_wait_asynccnt(0);
#else
  asm volatile("s_wait_asynccnt 0x0" ::: "memory");
#endif
#endif
}

// stage one (col,val) element into wave-private LDS slot (2 async b32 ops)
__device__ __forceinline__ void stage_edge(const int* __restrict__ gcol,
                                           const float* __restrict__ gval,
                                           int* lcol, float* lval,
                                           int idx, bool pred) {
  if (pred) {
#if defined(HAVE_ASYNC_LDS)
    __builtin_amdgcn_global_load_async_to_lds_b32(as_gbl(gcol + idx), as_lds(lcol), 0, 0);
    __builtin_amdgcn_global_load_async_to_lds_b32(as_gbl(gval + idx), as_lds(lval), 0, 0);
#else
    *lcol = gcol[idx];
    *lval = gval[idx];
#endif
  }
}

// ---------------------------------------------------------------------------
// init: acc = h_cur = concat(emb_user, emb_item); optionally zero h_next.
// ---------------------------------------------------------------------------
__global__ __launch_bounds__(BLOCK) void lgcn_init(
    const float4* __restrict__ eu, const float4* __restrict__ ei,
    float4* __restrict__ acc, float4* __restrict__ hcur,
    float4* __restrict__ hnext, int n4_user, int n4_total, int zero_next)
{
  int i = blockIdx.x * BLOCK + threadIdx.x;
  if (i >= n4_total) return;
  float4 v = (i < n4_user) ? eu[i] : ei[i - n4_user];
  acc[i]  = v;
  hcur[i] = v;
  if (zero_next) hnext[i] = make_float4(0.f, 0.f, 0.f, 0.f);
}

// ---------------------------------------------------------------------------
// CSR build: zero counts -> count -> scan(offsets) -> scatter permuted edges
// ---------------------------------------------------------------------------
__global__ __launch_bounds__(BLOCK) void lgcn_zero_i32(int* __restrict__ p, int n) {
  int i = blockIdx.x * BLOCK + threadIdx.x;
  if (i < n) p[i] = 0;
}

__global__ __launch_bounds__(BLOCK) void lgcn_count_rows(
    const int* __restrict__ rows, int* __restrict__ cnt, int nnz) {
  int i = blockIdx.x * BLOCK + threadIdx.x;
  if (i < nnz) atomicAdd(&cnt[rows[i]], 1);
}

// single-workgroup exclusive scan over N_NODES counts (147 chunks of 1024)
__global__ __launch_bounds__(1024) void lgcn_scan(
    const int* __restrict__ cnt, int* __restrict__ ptr,
    int* __restrict__ fill, int n)
{
  __shared__ int sdata[1024];
  __shared__ int carry;
  if (threadIdx.x == 0) carry = 0;
  __syncthreads();
  for (int base = 0; base < n; base += 1024) {
    int i = base + (int)threadIdx.x;
    int v = (i < n) ? cnt[i] : 0;
    sdata[threadIdx.x] = v;
    __syncthreads();
    for (int off = 1; off < 1024; off <<= 1) {
      int t = (threadIdx.x >= (unsigned)off) ? sdata[threadIdx.x - off] : 0;
      __syncthreads();
      sdata[threadIdx.x] += t;
      __syncthreads();
    }
    int excl = sdata[threadIdx.x] - v + carry;
    if (i < n) { ptr[i] = excl; fill[i] = excl; }
    __syncthreads();
    if (threadIdx.x == 1023) carry += sdata[1023];
    __syncthreads();
  }
  if (threadIdx.x == 0) ptr[n] = carry;   // == nnz
}

__global__ __launch_bounds__(BLOCK) void lgcn_scatter(
    const int* __restrict__ rows, const int* __restrict__ cols,
    const float* __restrict__ vals, int* __restrict__ fill,
    int* __restrict__ scol, float* __restrict__ sval, int nnz)
{
  int i = blockIdx.x * BLOCK + threadIdx.x;
  if (i >= nnz) return;
  int r = rows[i];
  int p = atomicAdd(&fill[r], 1);
  scol[p] = cols[i];
  sval[p] = vals[i];
}

// ---------------------------------------------------------------------------
// CSR SpMM (atomic-free): one wave32 per output row; lane owns float2 of 64
// dims. Edge (col,val) chunks double-buffered in LDS via async DMA.
// ---------------------------------------------------------------------------
__global__ __launch_bounds__(BLOCK) void lgcn_spmm_csr(
    const int* __restrict__ row_ptr, const int* __restrict__ scol,
    const float* __restrict__ sval, const float* __restrict__ x,
    float* __restrict__ y, int n_rows)
{
  __shared__ int   s_col[WPB][2][32];
  __shared__ float s_val[WPB][2][32];

  const int lane = threadIdx.x & 31;
  const int wave = threadIdx.x >> 5;
  const int row  = blockIdx.x * WPB + wave;
  if (row >= n_rows) return;          // wave-uniform; no barriers below

  const int start = row_ptr[row];
  const int end   = row_ptr[row + 1];
  const int nch   = (end - start + 31) >> 5;

  if (nch > 0) {                      // prime buffer 0 (2 async ops)
    int idx = start + lane;
    stage_edge(scol, sval, &s_col[wave][0][lane], &s_val[wave][0][lane],
               idx, idx < end);
  }

  float2 a = make_float2(0.f, 0.f);
  for (int c = 0; c < nch; ++c) {
    const bool more = (c + 1) < nch;
    if (more) {                       // prefetch next chunk into other buffer
      int idx = start + ((c + 1) << 5) + lane;
      stage_edge(scol, sval, &s_col[wave][(c + 1) & 1][lane],
                 &s_val[wave][(c + 1) & 1][lane], idx, idx < end);
      wait_async_le2();               // chunk c landed (in-order completion)
    } else {
      wait_async_le0();
    }
    const int cnt = min(32, end - start - (c << 5));
    const int*   cb = &s_col[wave][c & 1][0];
    const float* vb = &s_val[wave][c & 1][0];
    for (int j = 0; j < cnt; ++j) {
      const int   col = cb[j];        // uniform LDS broadcast
      const float v   = vb[j];
      const float2 xv = *(const float2*)(x + (col << 6) + (lane << 1));
      a.x = fmaf(v, xv.x, a.x);
      a.y = fmaf(v, xv.y, a.y);
    }
  }
  *(float2*)(y + (row << 6) + (lane << 1)) = a;   // plain store, no atomics
}

// ---------------------------------------------------------------------------
// COO SpMM fallback (atomic scatter) — used only if workspace is too small.
// ---------------------------------------------------------------------------
__global__ __launch_bounds__(BLOCK) void lgcn_spmm_coo(
    const float* __restrict__ vals, const int* __restrict__ rows,
    const int* __restrict__ cols,  const float* __restrict__ x,
    float* __restrict__ y, int nnz)
{
  __shared__ int   s_row[BLOCK];
  __shared__ int   s_col[BLOCK];
  __shared__ float s_val[BLOCK];

  const int tid   = threadIdx.x;
  const int lane  = tid & 31;
  const int wave  = tid >> 5;
  const int base  = blockIdx.x * BLOCK;
  const int count = min(BLOCK, nnz - base);
  const int e     = base + tid;

  if (tid < count) {
#if defined(HAVE_ASYNC_LDS)
    __builtin_amdgcn_global_load_async_to_lds_b32(as_gbl(rows + e), as_lds(&s_row[tid]), 0, 0);
    __builtin_amdgcn_global_load_async_to_lds_b32(as_gbl(cols + e), as_lds(&s_col[tid]), 0, 0);
    __builtin_amdgcn_global_load_async_to_lds_b32(as_gbl(vals + e), as_lds(&s_val[tid]), 0, 0);
#else
    s_row[tid] = rows[e]; s_col[tid] = cols[e]; s_val[tid] = vals[e];
#endif
  }
  wait_async_le0();
  __syncthreads();

  const int lo = wave * 32;
  const int hi = min(lo + 32, count);
  for (int j = lo; j < hi; ++j) {
    const int   r = s_row[j];
    const int   c = s_col[j];
    const float v = s_val[j];
    const float2 xv = *(const float2*)(x + (c << 6) + (lane << 1));
    float* yp = y + (r << 6) + (lane << 1);
    unsafeAtomicAdd(yp + 0, v * xv.x);
    unsafeAtomicAdd(yp + 1, v * xv.y);
  }
}

// ---------------------------------------------------------------------------
// accumulate variants
// ---------------------------------------------------------------------------
__global__ __launch_bounds__(BLOCK) void lgcn_accum_csr(
    float4* __restrict__ acc, const float4* __restrict__ h, int n4, int is_final)
{
  int i = blockIdx.x * BLOCK + threadIdx.x;
  if (i >= n4) return;
  float4 hv = h[i];
  float4 a  = acc[i];
  a.x += hv.x; a.y += hv.y; a.z += hv.z; a.w += hv.w;
  if (is_final) {
    const float s = 1.0f / (N_LAYERS + 1);
    a.x *= s; a.y *= s; a.z *= s; a.w *= s;
  }
  acc[i] = a;
}

__global__ __launch_bounds__(BLOCK) void lgcn_accum_coo(
    float4* __restrict__ acc, float4* __restrict__ hcur,
    float4* __restrict__ hnext, int n4, int is_final)
{
  int i = blockIdx.x * BLOCK + threadIdx.x;
  if (i >= n4) return;
  float4 h = hnext[i];
  float4 a = acc[i];
  a.x += h.x; a.y += h.y; a.z += h.z; a.w += h.w;
  if (is_final) {
    const float s = 1.0f / (N_LAYERS + 1);
    a.x *= s; a.y *= s; a.z *= s; a.w *= s;
    acc[i] = a;
  } else {
    acc[i]   = a;
    hcur[i]  = h;
    hnext[i] = make_float4(0.f, 0.f, 0.f, 0.f);
  }
}

// ---------------------------------------------------------------------------
extern "C" void kernel_launch(void* const* d_in, const int* in_sizes, int n_in,
                              void* d_out, int out_size, void* d_ws, size_t ws_size,
                              hipStream_t stream) {
  const float* emb_user = (const float*)d_in[0];
  const float* emb_item = (const float*)d_in[1];
  const float* adj_vals = (const float*)d_in[2];
  const int*   adj_rows = (const int*)d_in[3];
  const int*   adj_cols = (const int*)d_in[4];
  const int    nnz      = in_sizes[2];

  float* acc = (float*)d_out;                    // 150000*64 f32 (= output)
  const int n4_total = TOTAL / 4;                // 2,400,000
  const int n4_user  = (N_USERS * EMB) / 4;
  const int eb = (n4_total + BLOCK - 1) / BLOCK;
  const int sb = (nnz + BLOCK - 1) / BLOCK;      // 18750
  const int nb = (N_NODES + BLOCK - 1) / BLOCK;

  const size_t BYTES_H = (size_t)TOTAL * sizeof(float);           // 38.4 MB
  const size_t need_csr = 2 * BYTES_H + (size_t)nnz * 8           // scol+sval
                        + ((size_t)N_NODES + 1) * 4               // row_ptr
                        + (size_t)N_NODES * 4                     // row_fill
                        + (size_t)N_NODES * 4;                    // row_cnt

  char* w = (char*)d_ws;
  float* h_a = (float*)w;            w += BYTES_H;
  float* h_b = (float*)w;            w += BYTES_H;

  if (ws_size >= need_csr) {
    // ---------------- CSR path (atomic-free SpMM) ----------------
    int*   scol     = (int*)w;       w += (size_t)nnz * 4;
    float* sval     = (float*)w;     w += (size_t)nnz * 4;
    int*   row_ptr  = (int*)w;       w += ((size_t)N_NODES + 1) * 4;
    int*   row_fill = (int*)w;       w += (size_t)N_NODES * 4;
    int*   row_cnt  = (int*)w;

    lgcn_init<<<eb, BLOCK, 0, stream>>>(
        (const float4*)emb_user, (const float4*)emb_item,
        (float4*)acc, (float4*)h_a, (float4*)h_b, n4_user, n4_total, 0);

    lgcn_zero_i32<<<nb, BLOCK, 0, stream>>>(row_cnt, N_NODES);
    lgcn_count_rows<<<sb, BLOCK, 0, stream>>>(adj_rows, row_cnt, nnz);
    lgcn_scan<<<1, 1024, 0, stream>>>(row_cnt, row_ptr, row_fill, N_NODES);
    lgcn_scatter<<<sb, BLOCK, 0, stream>>>(adj_rows, adj_cols, adj_vals,
                                           row_fill, scol, sval, nnz);

    const int rb = (N_NODES + WPB - 1) / WPB;    // 18750 blocks, 1 wave/row
    float* x = h_a;
    float* y = h_b;
    for (int layer = 0; layer < N_LAYERS; ++layer) {
      lgcn_spmm_csr<<<rb, BLOCK, 0, stream>>>(row_ptr, scol, sval, x, y, N_NODES);
      lgcn_accum_csr<<<eb, BLOCK, 0, stream>>>((float4*)acc, (const float4*)y,
                                               n4_total, layer == N_LAYERS - 1);
      float* t = x; x = y; y = t;
    }
  } else {
    // ---------------- COO fallback (atomic scatter) ----------------
    float* hcur  = h_a;
    float* hnext = h_b;
    lgcn_init<<<eb, BLOCK, 0, stream>>>(
        (const float4*)emb_user, (const float4*)emb_item,
        (float4*)acc, (float4*)hcur, (float4*)hnext, n4_user, n4_total, 1);
    for (int layer = 0; layer < N_LAYERS; ++layer) {
      lgcn_spmm_coo<<<sb, BLOCK, 0, stream>>>(adj_vals, adj_rows, adj_cols,
                                              hcur, hnext, nnz);
      lgcn_accum_coo<<<eb, BLOCK, 0, stream>>>((float4*)acc, (float4*)hcur,
                                               (float4*)hnext, n4_total,
                                               layer == N_LAYERS - 1);
    }
  }
}